// decoder_70059506533024
// MI455X (gfx1250) — compile-verified
//
#include <hip/hip_runtime.h>
#include <hip/hip_bf16.h>
#include <math.h>

// ---------------------------------------------------------------------------
// Decoder step for MI455X (gfx1250, wave32).
// All GEMMs have M=64 -> exactly 4 row tiles of 16; V_WMMA_F32_16X16X4_F32.
// Vocab projection (201 MB out_w stream) uses ASYNC global->LDS double
// buffering per wave, synchronized with s_wait_asynccnt.
// ---------------------------------------------------------------------------

typedef __attribute__((ext_vector_type(2))) float v2f;
typedef __attribute__((ext_vector_type(8))) float v8f;
typedef int v4i __attribute__((vector_size(16)));
typedef __attribute__((address_space(1))) v4i* as1_v4i;
typedef __attribute__((address_space(3))) v4i* as3_v4i;

#define WINSZ 5
#define KW 11
#define NEGV -10000.0f

__device__ __forceinline__ float sigmoidf_(float x) { return 1.0f / (1.0f + __expf(-x)); }

// ---- async global -> LDS (B128) with builtin/asm fallback ------------------
__device__ __forceinline__ void async_b128(const float* g, float* l)
{
#if __has_builtin(__builtin_amdgcn_global_load_async_to_lds_b128)
    __builtin_amdgcn_global_load_async_to_lds_b128((as1_v4i)g, (as3_v4i)l, 0, 0);
#else
    unsigned lo = (unsigned)(size_t)(__attribute__((address_space(3))) void*)l;
    asm volatile("global_load_async_to_lds_b128 %0, %1, off"
                 :: "v"(lo), "v"(g) : "memory");
#endif
}

#if __has_builtin(__builtin_amdgcn_s_wait_asynccnt)
#define WAIT_ASYNC(n) __builtin_amdgcn_s_wait_asynccnt(n)
#else
#define WAIT_ASYNC(n) asm volatile("s_wait_asynccnt %0" :: "i"(n) : "memory")
#endif

// ---------------------------------------------------------------------------
// Generic M=64 GEMM: C[64,N] = act(A[64,K] @ B + bias)
//   TRANSB==0: B is [K,N] row-major (ldb=N)
//   TRANSB==1: B is [N,K] row-major (ldb=K)   (i.e. C = A @ B^T)
// One wave per 16-col tile, 4 accumulators for the 4 row tiles.
// OOB columns load from a clamped in-bounds address; WMMA keeps column n of C
// in lane r==n, so garbage stays in lanes whose stores are masked.
// ---------------------------------------------------------------------------
template <int TRANSB>
__global__ void __launch_bounds__(128)
wmma_gemm64(const float* __restrict__ A, int lda,
            const float* __restrict__ B, int ldb,
            const float* __restrict__ bias,
            float* __restrict__ C, int ldc,
            int N, int K, int act)
{
    const int lane  = threadIdx.x & 31;
    const int wave  = threadIdx.x >> 5;
    const int ntile = blockIdx.x * 4 + wave;
    const int n0    = ntile * 16;
    if (n0 >= N) return;                      // wave-uniform exit

    const int  r   = lane & 15;               // row-in-tile (A) / col-in-tile (B,C)
    const int  hi  = lane >> 4;               // half-wave: K offset 0 or 2
    const int  n   = n0 + r;
    const bool nok = (n < N);
    const int  ns  = nok ? n : (N - 1);       // clamped column for safe loads

    v8f acc0 = {}, acc1 = {}, acc2 = {}, acc3 = {};

#pragma unroll 4
    for (int k0 = 0; k0 < K; k0 += 4) {
        const int k = k0 + hi * 2;

        v2f bf;
        if constexpr (TRANSB) {
            const float* bp = B + (size_t)ns * ldb + k;
            bf.x = bp[0];
            bf.y = bp[1];
        } else {
            bf.x = B[(size_t)k * ldb + ns];
            bf.y = B[(size_t)(k + 1) * ldb + ns];
            if ((k0 & 63) == 0 && (k0 + 64) < K)      // uniform condition
                __builtin_prefetch(&B[(size_t)(k0 + 64) * ldb + ns], 0, 1);
        }

        const float* ap = A + (size_t)r * lda + k;
        v2f a0, a1, a2, a3;
        a0.x = ap[0];             a0.y = ap[1];
        a1.x = ap[16 * lda];      a1.y = ap[16 * lda + 1];
        a2.x = ap[32 * lda];      a2.y = ap[32 * lda + 1];
        a3.x = ap[48 * lda];      a3.y = ap[48 * lda + 1];

        acc0 = __builtin_amdgcn_wmma_f32_16x16x4_f32(false, a0, false, bf, (short)0, acc0, false, false);
        acc1 = __builtin_amdgcn_wmma_f32_16x16x4_f32(false, a1, false, bf, (short)0, acc1, false, false);
        acc2 = __builtin_amdgcn_wmma_f32_16x16x4_f32(false, a2, false, bf, (short)0, acc2, false, false);
        acc3 = __builtin_amdgcn_wmma_f32_16x16x4_f32(false, a3, false, bf, (short)0, acc3, false, false);
    }

    if (!nok) return;
    const float bv = bias ? bias[n] : 0.0f;

#pragma unroll
    for (int i = 0; i < 8; ++i) {
        const int m = i + hi * 8;             // C layout: VGPR i -> row i (+8 for hi half)
        float v0 = acc0[i] + bv;
        float v1 = acc1[i] + bv;
        float v2 = acc2[i] + bv;
        float v3 = acc3[i] + bv;
        if (act) { v0 = tanhf(v0); v1 = tanhf(v1); v2 = tanhf(v2); v3 = tanhf(v3); }
        C[(size_t)(m)      * ldc + n] = v0;
        C[(size_t)(m + 16) * ldc + n] = v1;
        C[(size_t)(m + 32) * ldc + n] = v2;
        C[(size_t)(m + 48) * ldc + n] = v3;
    }
}

// ---------------------------------------------------------------------------
// Vocab projection with async double buffering. Full 16-col tiles only.
// C[64, n0:n0+16] = A[64,K] @ B[K, n0:n0+16] + bias.  K % 40 == 0.
// Each wave: 2 LDS panels of 40x16 f32; 5 async B128 ops fill one panel
// (8 rows x 16 cols per op). Producer == consumer wave -> only
// s_wait_asynccnt is needed (in-order completion), no barriers.
// ---------------------------------------------------------------------------
#define CK 40

__device__ __forceinline__ void issue_panel(const float* __restrict__ B, int ldb,
                                            int n0, int k0, float* buf, int lane)
{
    const int rr = lane >> 2;        // 0..7
    const int cg = (lane & 3) * 4;   // 0,4,8,12
#pragma unroll
    for (int i = 0; i < 5; ++i) {
        const int row = i * 8 + rr;
        async_b128(B + (size_t)(k0 + row) * ldb + n0 + cg, buf + row * 16 + cg);
    }
}

__global__ void __launch_bounds__(128)
wmma_gemm64_async(const float* __restrict__ A, int lda,
                  const float* __restrict__ B, int ldb,
                  const float* __restrict__ bias,
                  float* __restrict__ C, int ldc,
                  int Ntiles, int K)
{
    __shared__ float ldsB[4][2][CK * 16];     // 20 KB

    const int lane  = threadIdx.x & 31;
    const int wave  = threadIdx.x >> 5;
    const int ntile = blockIdx.x * 4 + wave;
    if (ntile >= Ntiles) return;              // wave-uniform, no barriers used
    const int n0 = ntile * 16;

    const int r  = lane & 15;
    const int hi = lane >> 4;
    const int n  = n0 + r;

    v8f acc0 = {}, acc1 = {}, acc2 = {}, acc3 = {};

    const int chunks = K / CK;
    issue_panel(B, ldb, n0, 0, &ldsB[wave][0][0], lane);

    for (int c = 0; c < chunks; ++c) {
        if (c + 1 < chunks) {
            issue_panel(B, ldb, n0, (c + 1) * CK, &ldsB[wave][(c + 1) & 1][0], lane);
            WAIT_ASYNC(5);                    // previous panel complete
        } else {
            WAIT_ASYNC(0);
        }
        const float* panel = &ldsB[wave][c & 1][0];
        const int kbase = c * CK;

#pragma unroll
        for (int s = 0; s < CK / 4; ++s) {
            const int kl = s * 4 + hi * 2;
            v2f bf;
            bf.x = panel[kl * 16 + r];
            bf.y = panel[(kl + 1) * 16 + r];

            const float* ap = A + (size_t)r * lda + kbase + kl;
            v2f a0, a1, a2, a3;
            a0.x = ap[0];             a0.y = ap[1];
            a1.x = ap[16 * lda];      a1.y = ap[16 * lda + 1];
            a2.x = ap[32 * lda];      a2.y = ap[32 * lda + 1];
            a3.x = ap[48 * lda];      a3.y = ap[48 * lda + 1];

            acc0 = __builtin_amdgcn_wmma_f32_16x16x4_f32(false, a0, false, bf, (short)0, acc0, false, false);
            acc1 = __builtin_amdgcn_wmma_f32_16x16x4_f32(false, a1, false, bf, (short)0, acc1, false, false);
            acc2 = __builtin_amdgcn_wmma_f32_16x16x4_f32(false, a2, false, bf, (short)0, acc2, false, false);
            acc3 = __builtin_amdgcn_wmma_f32_16x16x4_f32(false, a3, false, bf, (short)0, acc3, false, false);
        }
    }

    const float bv = bias[n];
#pragma unroll
    for (int i = 0; i < 8; ++i) {
        const int m = i + hi * 8;
        C[(size_t)(m)      * ldc + n] = acc0[i] + bv;
        C[(size_t)(m + 16) * ldc + n] = acc1[i] + bv;
        C[(size_t)(m + 32) * ldc + n] = acc2[i] + bv;
        C[(size_t)(m + 48) * ldc + n] = acc3[i] + bv;
    }
}

// ---------------------------------------------------------------------------
// x[b, 0:300] = embed[dec_in[b]]; x[b, 300:1300] = attn_hidden[b]
// ---------------------------------------------------------------------------
__global__ void embed_concat(const int* __restrict__ dec_in,
                             const float* __restrict__ embed,
                             const float* __restrict__ attn_hidden,
                             float* __restrict__ x)
{
    int idx = blockIdx.x * blockDim.x + threadIdx.x;
    if (idx >= 64 * 1300) return;
    int b = idx / 1300, d = idx % 1300;
    float v;
    if (d < 300) {
        int tok = dec_in[b];
        v = embed[(size_t)tok * 300 + d];
    } else {
        v = attn_hidden[(size_t)b * 1000 + (d - 300)];
    }
    x[idx] = v;
}

// ---------------------------------------------------------------------------
// GRU gates: gi/gh are [2,64,1500] with biases already fused by the GEMM.
// ---------------------------------------------------------------------------
__global__ void gru_gate(const float* __restrict__ gi,
                         const float* __restrict__ gh,
                         const float* __restrict__ h0layer,   // [2,64,500]
                         float* __restrict__ xout)            // [64,1000]
{
    int idx = blockIdx.x * blockDim.x + threadIdx.x;
    if (idx >= 2 * 64 * 500) return;
    int dir = idx / 32000;
    int rem = idx % 32000;
    int b   = rem / 500;
    int j   = rem % 500;
    const float* gib = gi + ((size_t)dir * 64 + b) * 1500;
    const float* ghb = gh + ((size_t)dir * 64 + b) * 1500;
    float r  = sigmoidf_(gib[j]        + ghb[j]);
    float z  = sigmoidf_(gib[j + 500]  + ghb[j + 500]);
    float nn = tanhf    (gib[j + 1000] + r * ghb[j + 1000]);
    float hp = h0layer[((size_t)dir * 64 + b) * 500 + j];
    xout[(size_t)b * 1000 + dir * 500 + j] = (1.0f - z) * nn + z * hp;
}

// ---------------------------------------------------------------------------
// Per-batch local attention.  s_buf = tanh(ht @ Wp + b) precomputed.
// mask0 is redundant (m_w == valid because p1 <= lengths), so never loaded.
// ---------------------------------------------------------------------------
__global__ void __launch_bounds__(256)
attn_kernel(const float* __restrict__ s_buf,     // [64,1000]
            const float* __restrict__ ht,        // [64,1000]
            const float* __restrict__ enc_out,   // [64,1024,1000]
            const int*   __restrict__ lengths,   // [64]
            const float* __restrict__ Vp_w,      // [1000]
            const float* __restrict__ Vp_b,      // [1]
            float* __restrict__ cat)             // [64,2000]
{
    __shared__ float red[256];
    __shared__ float sht[1000];
    __shared__ float shs[KW][1000];
    __shared__ float sa[KW];
    __shared__ int   sp0, sp1;

    const int b   = blockIdx.x;
    const int tid = threadIdx.x;
    const int len = lengths[b];

    for (int d = tid; d < 1000; d += 256) sht[d] = ht[(size_t)b * 1000 + d];

    float part = 0.0f;
    for (int d = tid; d < 1000; d += 256) part += s_buf[(size_t)b * 1000 + d] * Vp_w[d];
    red[tid] = part; __syncthreads();
    for (int off = 128; off > 0; off >>= 1) {
        if (tid < off) red[tid] += red[tid + off];
        __syncthreads();
    }
    if (tid == 0) {
        float sc = sigmoidf_(red[0] + Vp_b[0]);
        int pt = (int)((float)len * sc);               // trunc, matches astype(int32)
        int p0 = pt - WINSZ;
        if (p0 < 0) p0 = 0;
        if (p0 > len - WINSZ) p0 = len - WINSZ;        // min(len-WIN, max(0, pt-WIN))
        int p1 = pt + 1 + WINSZ;
        if (p1 > len) p1 = len;
        sp0 = p0; sp1 = p1;
    }
    __syncthreads();
    const int p0 = sp0, p1 = sp1;

    for (int t = tid; t < KW * 1000; t += 256) {
        int k = t / 1000, d = t - k * 1000;
        int idx = p0 + k;
        int idxc = idx < 0 ? 0 : (idx > 1023 ? 1023 : idx);
        shs[k][d] = enc_out[((size_t)b * 1024 + idxc) * 1000 + d];
    }
    __syncthreads();

    for (int k = 0; k < KW; ++k) {
        float p = 0.0f;
        for (int d = tid; d < 1000; d += 256) p += sht[d] * shs[k][d];
        red[tid] = p; __syncthreads();
        for (int off = 128; off > 0; off >>= 1) {
            if (tid < off) red[tid] += red[tid + off];
            __syncthreads();
        }
        if (tid == 0) sa[k] = ((p0 + k) < p1) ? red[0] : NEGV;
        __syncthreads();
    }
    if (tid == 0) {
        float mx = sa[0];
        for (int k = 1; k < KW; ++k) mx = fmaxf(mx, sa[k]);
        float sum = 0.0f;
        for (int k = 0; k < KW; ++k) { sa[k] = __expf(sa[k] - mx); sum += sa[k]; }
        float inv = 1.0f / sum;
        for (int k = 0; k < KW; ++k) sa[k] *= inv;
    }
    __syncthreads();

    for (int d = tid; d < 1000; d += 256) {
        float c = 0.0f;
#pragma unroll
        for (int k = 0; k < KW; ++k) c += sa[k] * shs[k][d];
        cat[(size_t)b * 2000 + d]        = c;
        cat[(size_t)b * 2000 + 1000 + d] = sht[d];
    }
}

// ---------------------------------------------------------------------------
__global__ void __launch_bounds__(512)
lse_kernel(const float* __restrict__ y, float* __restrict__ lse)
{
    __shared__ float red[512];
    const int b = blockIdx.x, tid = threadIdx.x;
    const float* yb = y + (size_t)b * 50257;

    float mx = -3.4e38f;
    for (int i = tid; i < 50257; i += 512) mx = fmaxf(mx, yb[i]);
    red[tid] = mx; __syncthreads();
    for (int off = 256; off > 0; off >>= 1) {
        if (tid < off) red[tid] = fmaxf(red[tid], red[tid + off]);
        __syncthreads();
    }
    mx = red[0]; __syncthreads();

    float s = 0.0f;
    for (int i = tid; i < 50257; i += 512) s += __expf(yb[i] - mx);
    red[tid] = s; __syncthreads();
    for (int off = 256; off > 0; off >>= 1) {
        if (tid < off) red[tid] += red[tid + off];
        __syncthreads();
    }
    if (tid == 0) lse[b] = mx + __logf(red[0]);
}

__global__ void sub_lse(const float* __restrict__ y,
                        const float* __restrict__ lse,
                        float* __restrict__ out)
{
    size_t i = (size_t)blockIdx.x * blockDim.x + threadIdx.x;
    if (i >= (size_t)64 * 50257) return;
    int b = (int)(i / 50257);
    out[i] = y[i] - lse[b];
}

// ---------------------------------------------------------------------------
static inline void gemm64(const float* A, int lda, const float* B, int ldb, int transB,
                          const float* bias, float* C, int ldc, int N, int K, int act,
                          hipStream_t s)
{
    int ntiles = (N + 15) / 16;
    dim3 grid((ntiles + 3) / 4);
    if (transB)
        wmma_gemm64<1><<<grid, dim3(128), 0, s>>>(A, lda, B, ldb, bias, C, ldc, N, K, act);
    else
        wmma_gemm64<0><<<grid, dim3(128), 0, s>>>(A, lda, B, ldb, bias, C, ldc, N, K, act);
}

extern "C" void kernel_launch(void* const* d_in, const int* in_sizes, int n_in,
                              void* d_out, int out_size, void* d_ws, size_t ws_size,
                              hipStream_t stream)
{
    (void)in_sizes; (void)n_in; (void)out_size; (void)ws_size;

    const int*   dec_in  = (const int*)  d_in[0];
    const float* attn_h  = (const float*)d_in[1];
    const float* enc_out = (const float*)d_in[2];
    /* d_in[3] = mask0: provably redundant (m_w == valid); never read */
    const int*   lengths = (const int*)  d_in[4];
    const float* h0      = (const float*)d_in[5];
    const float* embed   = (const float*)d_in[6];
    const float* w_ih0   = (const float*)d_in[7];
    const float* w_hh0   = (const float*)d_in[8];
    const float* b_ih0   = (const float*)d_in[9];
    const float* b_hh0   = (const float*)d_in[10];
    const float* w_ih1   = (const float*)d_in[11];
    const float* w_hh1   = (const float*)d_in[12];
    const float* b_ih1   = (const float*)d_in[13];
    const float* b_hh1   = (const float*)d_in[14];
    const float* Wp_w    = (const float*)d_in[15];
    const float* Wp_b    = (const float*)d_in[16];
    const float* Vp_w    = (const float*)d_in[17];
    const float* Vp_b    = (const float*)d_in[18];
    const float* Wc_w    = (const float*)d_in[19];
    const float* Wc_b    = (const float*)d_in[20];
    const float* out_w   = (const float*)d_in[21];
    const float* out_b   = (const float*)d_in[22];
    float* out = (float*)d_out;

    // workspace layout (floats)
    float* ws   = (float*)d_ws;
    float* x    = ws;                 // 64*1300   =   83200
    float* gi   = x   + 83200;        // 2*64*1500 =  192000
    float* gh   = gi  + 192000;       // 2*64*1500 =  192000
    float* x1   = gh  + 192000;       // 64*1000   =   64000
    float* ht   = x1  + 64000;        // 64*1000   =   64000
    float* sbuf = ht  + 64000;        // 64*1000   =   64000
    float* cat  = sbuf+ 64000;        // 64*2000   =  128000
    float* hat  = cat + 128000;       // 64*1000   =   64000
    float* y    = hat + 64000;        // 64*50257  = 3216448
    float* lse  = y   + 3216448;      // 64

    // 1) x = [embed[tok], attn_hidden]
    embed_concat<<<dim3((64 * 1300 + 255) / 256), dim3(256), 0, stream>>>(dec_in, embed, attn_h, x);

    // 2) GRU layer 0
    for (int d = 0; d < 2; ++d) {
        gemm64(x, 1300, w_ih0 + (size_t)d * 1500 * 1300, 1300, 1, b_ih0 + d * 1500,
               gi + (size_t)d * 64 * 1500, 1500, 1500, 1300, 0, stream);
        gemm64(h0 + (size_t)d * 64 * 500, 500, w_hh0 + (size_t)d * 1500 * 500, 500, 1,
               b_hh0 + d * 1500, gh + (size_t)d * 64 * 1500, 1500, 1500, 500, 0, stream);
    }
    gru_gate<<<dim3((64000 + 255) / 256), dim3(256), 0, stream>>>(gi, gh, h0, x1);

    // 3) GRU layer 1
    for (int d = 0; d < 2; ++d) {
        gemm64(x1, 1000, w_ih1 + (size_t)d * 1500 * 1000, 1000, 1, b_ih1 + d * 1500,
               gi + (size_t)d * 64 * 1500, 1500, 1500, 1000, 0, stream);
        gemm64(h0 + (size_t)(2 + d) * 64 * 500, 500, w_hh1 + (size_t)d * 1500 * 500, 500, 1,
               b_hh1 + d * 1500, gh + (size_t)d * 64 * 1500, 1500, 1500, 500, 0, stream);
    }
    gru_gate<<<dim3((64000 + 255) / 256), dim3(256), 0, stream>>>(gi, gh, h0 + 2 * 64 * 500, ht);

    // 4) s = tanh(ht @ Wp_w + Wp_b)
    gemm64(ht, 1000, Wp_w, 1000, 0, Wp_b, sbuf, 1000, 1000, 1000, 1, stream);

    // 5) attention -> cat = [c, ht]
    attn_kernel<<<dim3(64), dim3(256), 0, stream>>>(sbuf, ht, enc_out, lengths, Vp_w, Vp_b, cat);

    // 6) h_attn = tanh(cat @ Wc_w + Wc_b)
    gemm64(cat, 2000, Wc_w, 1000, 0, Wc_b, hat, 1000, 1000, 2000, 1, stream);

    // 7) y = h_attn @ out_w + out_b
    //    full tiles via async double-buffered kernel; 1 leftover column via generic
    {
        const int NtilesFull = 50257 / 16;            // 3141 tiles = 50256 cols
        dim3 grid((NtilesFull + 3) / 4);
        wmma_gemm64_async<<<grid, dim3(128), 0, stream>>>(hat, 1000, out_w, 50257,
                                                          out_b, y, 50257, NtilesFull, 1000);
        const int nrem = 50257 - NtilesFull * 16;     // 1
        gemm64(hat, 1000, out_w + NtilesFull * 16, 50257, 0, out_b + NtilesFull * 16,
               y + NtilesFull * 16, 50257, nrem, 1000, 0, stream);
    }

    // 8) log_softmax
    lse_kernel<<<dim3(64), dim3(512), 0, stream>>>(y, lse);
    sub_lse<<<dim3((int)(((size_t)64 * 50257 + 255) / 256)), dim3(256), 0, stream>>>(y, lse, out);
}